// ScoringFunction_84997402787992
// MI455X (gfx1250) — compile-verified
//
#include <hip/hip_runtime.h>
#include <math.h>
#include <stdint.h>

typedef float v2f __attribute__((ext_vector_type(2)));
typedef float v8f __attribute__((ext_vector_type(8)));

#define WAVE 32
#define KCHUNK 128            // k-depth staged in LDS per chunk
#define STR 132               // LDS column stride in floats: %4==0 (align), %64==4 (bank skew)

// -------------------------------------------------------------------------
// Kernel 1: U = h @ W  (N x D) @ (D x D), fp32 WMMA 16x16x4.
// Block = 256 threads (8 waves). Block owns one 64-wide N tile; W is staged
// into LDS column-major (transposed in flight) with GLOBAL_LOAD_ASYNC_TO_LDS_B32
// scatter (ASYNCcnt path), two 128-deep k chunks. Each wave computes a
// 64(M)x64(N) tile: 16 v8f accumulators; per k-step: 4 global b64 (A)
// + 4 ds b64 (B, pair-contiguous) + 16 v_wmma_f32_16x16x4_f32.
// Fragment layouts (ISA 7.12.2):
//   A (16x4 f32): lane(lo,hi) holds A[lo][k+2*hi], A[lo][k+2*hi+1]
//   B (4x16 f32): lane(lo,hi) holds B[2*hi][lo],   B[2*hi+1][lo]
//   C (16x16 f32): vgpr r -> (row r + 8*hi, col lo)
// -------------------------------------------------------------------------
__global__ __launch_bounds__(256) void hyperedge_gemm_hW(
    const float* __restrict__ h, const float* __restrict__ W,
    float* __restrict__ U, int N, int D) {
  __shared__ float wtile[64 * STR];   // 33792 B, column-major W chunk

  const int nTiles = D / 64;
  const int nTile  = blockIdx.x % nTiles;
  const int mBlock = blockIdx.x / nTiles;
  const int nBase  = nTile * 64;
  const int tid    = threadIdx.x;
  const int lane   = tid & (WAVE - 1);
  const int lo     = lane & 15;
  const int hi     = lane >> 4;

  // Each wave owns 64 M-rows; block covers 512 rows.
  const int mBase = (mBlock * 8 + (tid / WAVE)) * 64;

  // A row pointers (clamped for the tail tile; stores are guarded below).
  const float* ap[4];
#pragma unroll
  for (int s = 0; s < 4; ++s) {
    int row = mBase + 16 * s + lo;
    if (row >= N) row = N - 1;
    ap[s] = h + (size_t)row * D + 2 * hi;
  }

  v8f acc[4][4] = {};

  for (int kc = 0; kc < D; kc += KCHUNK) {
    __syncthreads();   // previous chunk's LDS reads complete before overwrite

    // ---- stage W[kc..kc+127][nBase..nBase+63] into LDS, column-major ----
    // Async DMA scatter: per-lane global addr (coalesced along columns) ->
    // per-lane LDS addr (transposed). Tracked by ASYNCcnt, no VGPR round-trip.
    for (int idx = tid; idx < 64 * KCHUNK; idx += blockDim.x) {
      const int c = idx & 63;            // column (fast -> coalesced global)
      const int k = idx >> 6;            // 0..127
      const float*   gsrc = W + (size_t)(kc + k) * D + nBase + c;
      const uint32_t ldst = (uint32_t)(size_t)(wtile + c * STR + k);
      asm volatile("global_load_async_to_lds_b32 %0, %1, off"
                   :: "v"(ldst), "v"(gsrc)
                   : "memory");
    }
    asm volatile("s_wait_asynccnt 0x0" ::: "memory");
    __syncthreads();

    // prefetch next k-chunk of the A rows (global_prefetch_b8)
    if (kc + KCHUNK < D) {
#pragma unroll
      for (int s = 0; s < 4; ++s) __builtin_prefetch(ap[s] + kc + KCHUNK, 0, 0);
    }

    for (int k = 0; k < KCHUNK; k += 4) {
      v2f a[4];
#pragma unroll
      for (int s = 0; s < 4; ++s) a[s] = *(const v2f*)(ap[s] + kc + k);

      // B fragments: pair (rows k+2*hi, k+2*hi+1) contiguous -> ds b64 loads
      const float* lp = wtile + lo * STR + k + 2 * hi;
      v2f b[4];
      b[0] = *(const v2f*)(lp);
      b[1] = *(const v2f*)(lp + 16 * STR);
      b[2] = *(const v2f*)(lp + 32 * STR);
      b[3] = *(const v2f*)(lp + 48 * STR);

#pragma unroll
      for (int s = 0; s < 4; ++s)
#pragma unroll
        for (int t = 0; t < 4; ++t)
          acc[s][t] = __builtin_amdgcn_wmma_f32_16x16x4_f32(
              false, a[s], false, b[t], (short)0, acc[s][t], false, false);
    }
  }

  // ---- epilogue: scatter C tiles to U ----
#pragma unroll
  for (int s = 0; s < 4; ++s) {
#pragma unroll
    for (int r = 0; r < 8; ++r) {
      const int row = mBase + 16 * s + r + 8 * hi;
      if (row < N) {
        float* __restrict__ urow = U + (size_t)row * D + nBase + lo;
        urow[0]  = acc[s][0][r];
        urow[16] = acc[s][1][r];
        urow[32] = acc[s][2][r];
        urow[48] = acc[s][3][r];
      }
    }
  }
}

// -------------------------------------------------------------------------
// Kernel 2: per-edge scoring. One wave32 per hyperedge, lane owns 8 floats.
// second = h1.U0 + h2.(U0+U1) + h3.(U0+U1+U2); first = 0.25*(sum hi).b
// -------------------------------------------------------------------------
__device__ __forceinline__ float4 ld4(const float* p) { return *(const float4*)p; }
__device__ __forceinline__ float4 add4(float4 a, float4 b) {
  return make_float4(a.x + b.x, a.y + b.y, a.z + b.z, a.w + b.w);
}
__device__ __forceinline__ float dot4(float4 a, float4 b) {
  return a.x * b.x + a.y * b.y + a.z * b.z + a.w * b.w;
}

__global__ __launch_bounds__(256) void hyperedge_score(
    const float* __restrict__ h, const int* __restrict__ x,
    const float* __restrict__ bvec, const float* __restrict__ cscalar,
    const float* __restrict__ U, float* __restrict__ out, int E, int D) {
  const int wavesPerBlock = blockDim.x / WAVE;
  const int e = blockIdx.x * wavesPerBlock + (threadIdx.x / WAVE);
  if (e >= E) return;
  const int lane = threadIdx.x & (WAVE - 1);
  const int base = lane * (D / WAVE);         // 8 floats per lane (D=256)

  const int i0 = x[(size_t)e * 4 + 0];
  const int i1 = x[(size_t)e * 4 + 1];
  const int i2 = x[(size_t)e * 4 + 2];
  const int i3 = x[(size_t)e * 4 + 3];

  const float* h0p = h + (size_t)i0 * D + base;
  const float* h1p = h + (size_t)i1 * D + base;
  const float* h2p = h + (size_t)i2 * D + base;
  const float* h3p = h + (size_t)i3 * D + base;
  const float* u0p = U + (size_t)i0 * D + base;
  const float* u1p = U + (size_t)i1 * D + base;
  const float* u2p = U + (size_t)i2 * D + base;

  float4 h0a = ld4(h0p),     h0b = ld4(h0p + 4);
  float4 h1a = ld4(h1p),     h1b = ld4(h1p + 4);
  float4 h2a = ld4(h2p),     h2b = ld4(h2p + 4);
  float4 h3a = ld4(h3p),     h3b = ld4(h3p + 4);
  float4 u0a = ld4(u0p),     u0b = ld4(u0p + 4);
  float4 u1a = ld4(u1p),     u1b = ld4(u1p + 4);
  float4 u2a = ld4(u2p),     u2b = ld4(u2p + 4);
  float4 ba  = ld4(bvec + base), bb = ld4(bvec + base + 4);

  // first term: 0.25 * (h0+h1+h2+h3) . b
  float4 sa = add4(add4(h0a, h1a), add4(h2a, h3a));
  float4 sb = add4(add4(h0b, h1b), add4(h2b, h3b));
  float first = dot4(sa, ba) + dot4(sb, bb);

  // prefix sums of U rows
  float4 p0a = u0a,            p0b = u0b;             // U0
  float4 p1a = add4(p0a, u1a), p1b = add4(p0b, u1b);  // U0+U1
  float4 p2a = add4(p1a, u2a), p2b = add4(p1b, u2b);  // U0+U1+U2

  float second = dot4(h1a, p0a) + dot4(h1b, p0b)
               + dot4(h2a, p1a) + dot4(h2b, p1b)
               + dot4(h3a, p2a) + dot4(h3b, p2b);

  float partial = 0.25f * first + second;
#pragma unroll
  for (int off = WAVE / 2; off > 0; off >>= 1)
    partial += __shfl_xor(partial, off, WAVE);

  if (lane == 0) {
    const float score = partial + cscalar[0];
    out[e] = 1.0f / (1.0f + __expf(-score));
  }
}

// -------------------------------------------------------------------------
extern "C" void kernel_launch(void* const* d_in, const int* in_sizes, int n_in,
                              void* d_out, int out_size, void* d_ws, size_t ws_size,
                              hipStream_t stream) {
  const float* h    = (const float*)d_in[0];
  const int*   x    = (const int*)  d_in[1];
  const float* bvec = (const float*)d_in[2];
  const float* W    = (const float*)d_in[3];
  const float* c    = (const float*)d_in[4];
  float* out = (float*)d_out;

  const int D = in_sizes[2];            // b has D elements (256)
  const int N = in_sizes[0] / D;        // 50000
  const int E = in_sizes[1] / 4;        // 100000

  float* U = (float*)d_ws;              // N*D floats = 51.2 MB scratch

  const int threads = 256;
  const int wavesPerBlock = threads / WAVE;

  // Kernel 1: U = h @ W   (block covers 512 M-rows x 64 N-cols)
  const int nTiles  = D / 64;
  const int mBlocks = (N + 511) / 512;
  const int grid1   = mBlocks * nTiles;
  hyperedge_gemm_hW<<<grid1, threads, 0, stream>>>(h, W, U, N, D);

  // Kernel 2: per-edge scores
  const int grid2 = (E + wavesPerBlock - 1) / wavesPerBlock;
  hyperedge_score<<<grid2, threads, 0, stream>>>(h, x, bvec, c, U, out, E, D);
}